// GAE_51780125720795
// MI455X (gfx1250) — compile-verified
//
#include <hip/hip_runtime.h>
#include <hip/hip_bf16.h>

typedef __attribute__((ext_vector_type(2))) float v2f;
typedef __attribute__((ext_vector_type(4))) float v4f;
typedef __attribute__((ext_vector_type(8))) float v8f;

#define NNODES 100000
#define NEDGES 1600000

// ---------------------------------------------------------------------------
// Wave-level fp32 WMMA GEMM: Y[M x C] = relu(X[M x C] @ W[C x C]^T + b)
// One wave computes a 16x16 tile of Y. K = C (multiple of 4).
// A (16x4 f32): lane r=lane&15 -> row r; half=lane>>4 picks K pair {2h, 2h+1}.
// B (4x16 f32) = W^T: same addressing pattern against weight row r.
// D (16x16 f32, 8 VGPRs): acc[i] -> row (i + 8*half), col (lane&15).
// ---------------------------------------------------------------------------
__global__ __launch_bounds__(256) void sage_proj_gemm_relu(
    const float* __restrict__ X, const float* __restrict__ W,
    const float* __restrict__ bias, float* __restrict__ Y, int C) {
  const int lane = threadIdx.x & 31;
  const int wave = blockIdx.x * (blockDim.x >> 5) + (threadIdx.x >> 5);
  const int tilesN = C >> 4;
  const int numTiles = (NNODES >> 4) * tilesN;
  if (wave >= numTiles) return;  // wave-uniform, EXEC stays all-1s inside
  const int tm = wave / tilesN;
  const int tn = wave - tm * tilesN;
  const int r = lane & 15;
  const int half = lane >> 4;

  const float* arow = X + (size_t)(tm * 16 + r) * C + 2 * half;
  const float* brow = W + (size_t)(tn * 16 + r) * C + 2 * half;

  v8f acc = {};
  for (int k = 0; k < C; k += 4) {
    v2f a = *(const v2f*)(arow + k);
    v2f b = *(const v2f*)(brow + k);
    acc = __builtin_amdgcn_wmma_f32_16x16x4_f32(
        false, a, false, b, (short)0, acc, false, false);
  }

  const float bv = bias[tn * 16 + r];
  float* yout = Y + (size_t)(tm * 16 + 8 * half) * C + tn * 16 + r;
#pragma unroll
  for (int i = 0; i < 8; ++i) {
    float v = acc[i] + bv;
    yout[(size_t)i * C] = v > 0.0f ? v : 0.0f;
  }
}

// ---------------------------------------------------------------------------
// Y[M x Cout] = AGG @ Wl^T + bl + X @ Wr^T   (optional relu)
// Two K-loops accumulate into the same 16x16 WMMA tile.
// ---------------------------------------------------------------------------
__global__ __launch_bounds__(256) void sage_out_gemm(
    const float* __restrict__ AGG, const float* __restrict__ X,
    const float* __restrict__ Wl, const float* __restrict__ Wr,
    const float* __restrict__ bias, float* __restrict__ Y,
    int Cin, int Cout, int do_relu) {
  const int lane = threadIdx.x & 31;
  const int wave = blockIdx.x * (blockDim.x >> 5) + (threadIdx.x >> 5);
  const int tilesN = Cout >> 4;
  const int numTiles = (NNODES >> 4) * tilesN;
  if (wave >= numTiles) return;
  const int tm = wave / tilesN;
  const int tn = wave - tm * tilesN;
  const int r = lane & 15;
  const int half = lane >> 4;

  const float* arow = AGG + (size_t)(tm * 16 + r) * Cin + 2 * half;
  const float* xrow = X + (size_t)(tm * 16 + r) * Cin + 2 * half;
  const float* lrow = Wl + (size_t)(tn * 16 + r) * Cin + 2 * half;
  const float* rrow = Wr + (size_t)(tn * 16 + r) * Cin + 2 * half;

  v8f acc = {};
  for (int k = 0; k < Cin; k += 4) {
    v2f a = *(const v2f*)(arow + k);
    v2f b = *(const v2f*)(lrow + k);
    acc = __builtin_amdgcn_wmma_f32_16x16x4_f32(
        false, a, false, b, (short)0, acc, false, false);
  }
  for (int k = 0; k < Cin; k += 4) {
    v2f a = *(const v2f*)(xrow + k);
    v2f b = *(const v2f*)(rrow + k);
    acc = __builtin_amdgcn_wmma_f32_16x16x4_f32(
        false, a, false, b, (short)0, acc, false, false);
  }

  const float bv = bias[tn * 16 + r];
  float* yout = Y + (size_t)(tm * 16 + 8 * half) * Cout + tn * 16 + r;
#pragma unroll
  for (int i = 0; i < 8; ++i) {
    float v = acc[i] + bv;
    if (do_relu) v = v > 0.0f ? v : 0.0f;
    yout[(size_t)i * Cout] = v;
  }
}

// ---------------------------------------------------------------------------
// Scatter-add of messages: AGG[dst] += H[src], one thread per (edge, 4 chans).
// b128 gather load + 4x global_atomic_add_f32 (hits mostly in L2: the AGG
// array is <= 51 MB vs 192 MB L2).
// ---------------------------------------------------------------------------
__global__ __launch_bounds__(256) void sage_scatter_add(
    const float* __restrict__ H, const int* __restrict__ srcIdx,
    const int* __restrict__ dstIdx, float* __restrict__ AGG,
    int C, int chunkShift) {
  long long tid = (long long)blockIdx.x * blockDim.x + threadIdx.x;
  long long total = (long long)NEDGES << chunkShift;
  if (tid >= total) return;
  int e = (int)(tid >> chunkShift);
  int c4 = (int)(tid & ((1 << chunkShift) - 1)) << 2;
  int s = srcIdx[e];
  int d = dstIdx[e];
  v4f m = *(const v4f*)(H + (size_t)s * C + c4);
  float* dp = AGG + (size_t)d * C + c4;
  unsafeAtomicAdd(dp + 0, m.x);
  unsafeAtomicAdd(dp + 1, m.y);
  unsafeAtomicAdd(dp + 2, m.z);
  unsafeAtomicAdd(dp + 3, m.w);
}

// ---------------------------------------------------------------------------
// Host-side layer driver
// ---------------------------------------------------------------------------
static void sage_layer(const float* X, int Cin, int Cout,
                       const float* Wp, const float* bp,
                       const float* Wl, const float* bl, const float* Wr,
                       const int* src, const int* dst,
                       float* H, float* AGG, float* Y, int relu,
                       hipStream_t stream) {
  {  // H = relu(X @ Wp^T + bp)
    int tiles = (NNODES >> 4) * (Cin >> 4);
    int grid = (tiles * 32 + 255) / 256;
    sage_proj_gemm_relu<<<grid, 256, 0, stream>>>(X, Wp, bp, H, Cin);
  }
  hipMemsetAsync(AGG, 0, (size_t)NNODES * Cin * sizeof(float), stream);
  {  // AGG[dst] += H[src]
    int chunkShift = (Cin == 128) ? 5 : (Cin == 64) ? 4 : 3;  // log2(C/4)
    long long total = (long long)NEDGES << chunkShift;
    int grid = (int)((total + 255) / 256);
    sage_scatter_add<<<grid, 256, 0, stream>>>(H, src, dst, AGG, Cin,
                                               chunkShift);
  }
  {  // Y = AGG @ Wl^T + bl + X @ Wr^T  (opt relu)
    int tiles = (NNODES >> 4) * (Cout >> 4);
    int grid = (tiles * 32 + 255) / 256;
    sage_out_gemm<<<grid, 256, 0, stream>>>(AGG, X, Wl, Wr, bl, Y, Cin, Cout,
                                            relu);
  }
}

extern "C" void kernel_launch(void* const* d_in, const int* in_sizes, int n_in,
                              void* d_out, int out_size, void* d_ws,
                              size_t ws_size, hipStream_t stream) {
  const float* x = (const float*)d_in[0];
  const int* ei = (const int*)d_in[1];
  const int* src = ei;            // edge_index[0] : message sources
  const int* dst = ei + NEDGES;   // edge_index[1] : aggregation targets

  // Params: for each layer 5 tensors: Wp, bp, Wl, bl, Wr
  const float* e1p[5];
  const float* e2p[5];
  const float* d1p[5];
  const float* d2p[5];
  for (int i = 0; i < 5; ++i) {
    e1p[i] = (const float*)d_in[2 + i];
    e2p[i] = (const float*)d_in[7 + i];
    d1p[i] = (const float*)d_in[12 + i];
    d2p[i] = (const float*)d_in[17 + i];
  }

  // Workspace: 3 buffers of N x 128 floats (ping-pong H / AGG / activation)
  float* W0 = (float*)d_ws;
  float* W1 = W0 + (size_t)NNODES * 128;
  float* W2 = W1 + (size_t)NNODES * 128;

  float* x_rec = (float*)d_out;                        // [N, 64]
  float* z = (float*)d_out + (size_t)NNODES * 64;      // [N, 32]

  // e1: 64 -> 128, relu around sage_conv
  sage_layer(x, 64, 128, e1p[0], e1p[1], e1p[2], e1p[3], e1p[4],
             src, dst, W0, W1, W2, 1, stream);
  // e2: 128 -> 32 (latent z, no relu)
  sage_layer(W2, 128, 32, e2p[0], e2p[1], e2p[2], e2p[3], e2p[4],
             src, dst, W0, W1, z, 0, stream);
  // d1: 32 -> 128, relu
  sage_layer(z, 32, 128, d1p[0], d1p[1], d1p[2], d1p[3], d1p[4],
             src, dst, W0, W1, W2, 1, stream);
  // d2: 128 -> 64 (reconstruction, no relu)
  sage_layer(W2, 128, 64, d2p[0], d2p[1], d2p[2], d2p[3], d2p[4],
             src, dst, W0, W1, x_rec, 0, stream);
}